// mulitiattention_14851996909842
// MI455X (gfx1250) — compile-verified
//
#include <hip/hip_runtime.h>

#define B_DIM 16
#define G_DIM 1708
#define H_DIM 5
#define GPAD  1712          // G rounded up to multiple of 16
#define BLOCK 128
#define WPB   (BLOCK / 32)  // waves per block (wave32)
#define LOG2E 1.44269504088896340736f

typedef __attribute__((ext_vector_type(2))) float v2f;
typedef __attribute__((ext_vector_type(8))) float v8f;

#if __has_builtin(__builtin_amdgcn_exp2f)
#define EXP2F(v) __builtin_amdgcn_exp2f(v)
#else
#define EXP2F(v) exp2f(v)
#endif

// Rank-1 softmax attention, fully fused: no (B,G,G) materialization.
// Per wave: one 16-wide i-tile; scores computed as 16x16 outer-product tiles
// via V_WMMA_F32_16X16X4_F32 (K=1 used), exp via native v_exp_f32.
__launch_bounds__(BLOCK)
__global__ void mha_rank1_wmma_kernel(const float* __restrict__ x,
                                      const float* __restrict__ WQ,
                                      const float* __restrict__ WK,
                                      const float* __restrict__ WV,
                                      const float* __restrict__ W0,
                                      float* __restrict__ out)
{
    __shared__ __align__(16) float x_lds[GPAD];
    __shared__ __align__(16) float k_lds[GPAD];  // x*WK*log2(e), zero padded
    __shared__ __align__(16) float v_lds[GPAD];  // x*WV, zero padded

    const int b    = blockIdx.x;
    const int tid  = threadIdx.x;
    const int wave = tid >> 5;
    const int lane = tid & 31;
    const int ln   = lane & 15;   // N index (i within tile)
    const int hi   = lane >> 4;   // M half: rows hi*8 .. hi*8+7

    const int  tile   = blockIdx.y * WPB + wave;
    const bool tValid = (tile * 16) < G_DIM;     // uniform per wave
    const int  i      = tile * 16 + ln;
    const int  iC     = (i < G_DIM) ? i : (G_DIM - 1);

    for (int j = tid; j < GPAD; j += BLOCK)
        x_lds[j] = (j < G_DIM) ? x[b * G_DIM + j] : 0.0f;
    __syncthreads();

    float out_acc = 0.0f;

    for (int h = 0; h < H_DIM; ++h) {
        const float* __restrict__ wk = WK + h * G_DIM;
        const float* __restrict__ wv = WV + h * G_DIM;
        for (int j = tid; j < GPAD; j += BLOCK) {
            const float xv = x_lds[j];
            const float kk = (j < G_DIM) ? wk[j] : 0.0f;
            const float vv = (j < G_DIM) ? wv[j] : 0.0f;
            k_lds[j] = xv * kk * LOG2E;  // exp(score) == exp2(t * k_lds)
            v_lds[j] = xv * vv;
        }
        __syncthreads();

        if (tValid) {
            // B operand: row K=0 holds t_i in lanes 0..15; all else zero.
            const float t = (lane < 16) ? x_lds[iC] * WQ[h * G_DIM + iC] : 0.0f;
            v2f bm; bm[0] = t; bm[1] = 0.0f;

            float Dsum = 0.0f, Nsum = 0.0f;

            int jb = 0;
            for (; jb + 16 <= G_DIM; jb += 16) {
                // A operand: column K=0 holds k_j (M=lane) in lanes 0..15.
                v2f am; am[0] = (lane < 16) ? k_lds[jb + ln] : 0.0f; am[1] = 0.0f;
                v8f c = {};
                v8f d = __builtin_amdgcn_wmma_f32_16x16x4_f32(
                    false, am, false, bm, (short)0, c, false, false);
                // lane holds D[m][n] with n = ln, m = r + 8*hi
                const float4* vp =
                    reinterpret_cast<const float4*>(v_lds + jb + (hi << 3));
                const float4 va = vp[0], vb = vp[1];
                const float vv[8] = {va.x, va.y, va.z, va.w,
                                     vb.x, vb.y, vb.z, vb.w};
#pragma unroll
                for (int r = 0; r < 8; ++r) {
                    const float e = EXP2F(d[r]);
                    Dsum += e;
                    Nsum += e * vv[r];
                }
            }
            {   // tail chunk: jb = 1696, j valid for 12 of 16 rows
                v2f am; am[0] = (lane < 16) ? k_lds[jb + ln] : 0.0f; am[1] = 0.0f;
                v8f c = {};
                v8f d = __builtin_amdgcn_wmma_f32_16x16x4_f32(
                    false, am, false, bm, (short)0, c, false, false);
                const float4* vp =
                    reinterpret_cast<const float4*>(v_lds + jb + (hi << 3));
                const float4 va = vp[0], vb = vp[1];
                const float vv[8] = {va.x, va.y, va.z, va.w,
                                     vb.x, vb.y, vb.z, vb.w};
#pragma unroll
                for (int r = 0; r < 8; ++r) {
                    const bool ok = (jb + (hi << 3) + r) < G_DIM;
                    const float e = ok ? EXP2F(d[r]) : 0.0f;
                    Dsum += e;
                    Nsum += e * vv[r];
                }
            }

            // merge M-halves: lanes l and l^16 hold complementary j-partials
            Dsum += __shfl_xor(Dsum, 16, 32);
            Nsum += __shfl_xor(Nsum, 16, 32);

            if (lane < 16 && i < G_DIM) {
                const float eii = EXP2F(t * k_lds[i]);  // diagonal correction
                const float num = Nsum - eii * v_lds[i];
                out_acc += W0[h] * num / Dsum;
            }
        }
        __syncthreads();  // protect k_lds/v_lds before next head's fill
    }

    if (tValid && lane < 16 && i < G_DIM)
        out[b * G_DIM + i] = out_acc;
}

extern "C" void kernel_launch(void* const* d_in, const int* in_sizes, int n_in,
                              void* d_out, int out_size, void* d_ws, size_t ws_size,
                              hipStream_t stream) {
    (void)in_sizes; (void)n_in; (void)out_size; (void)d_ws; (void)ws_size;
    const float* x  = (const float*)d_in[0];
    const float* WQ = (const float*)d_in[1];
    const float* WK = (const float*)d_in[2];
    const float* WV = (const float*)d_in[3];
    const float* W0 = (const float*)d_in[4];
    float* out = (float*)d_out;

    const int numTiles = (G_DIM + 15) / 16;          // 107 i-tiles
    const int groups   = (numTiles + WPB - 1) / WPB; // 27 -> grid 16 x 27
    dim3 grid(B_DIM, groups);
    hipLaunchKernelGGL(mha_rank1_wmma_kernel, grid, dim3(BLOCK), 0, stream,
                       x, WQ, WK, WV, W0, out);
}